// DeepSSM_Net_63831803953438
// MI455X (gfx1250) — compile-verified
//
#include <hip/hip_runtime.h>
#include <hip/hip_bf16.h>

typedef float v2f __attribute__((ext_vector_type(2)));
typedef float v8f __attribute__((ext_vector_type(8)));

// ---------------- problem constants ----------------
#define BATCH   128
#define NPTS    65536
#define NPOINT  10
#define NSAMPLE 8
#define RAD2    0.04f
#define EPSV    1e-5f

// ---------------- workspace layout (float offsets) ----------------
#define OFF_CENTERS 0u                         // 128*10*3  = 3840
#define OFF_GROUPED 4096u                      // 128*10*8*3 = 30720
#define OFF_T1      36864u                     // 128*8*8*10 = 81920
#define OFF_U1      (OFF_T1 + 81920u)
#define OFF_T2      (OFF_U1 + 81920u)
#define OFF_U2      (OFF_T2 + 81920u)
#define OFF_T3      (OFF_U2 + 81920u)          // 128*16*8*10 = 163840
#define OFF_FEAT    (OFF_T3 + 163840u)         // 128*160 = 20480
#define OFF_H       (OFF_FEAT + 20480u)        // 128*128 = 16384
#define OFF_SUMS    (OFF_H + 16384u)           // 320 floats of reduction state
#define SUMS_FLOATS 320

// =====================================================================
// 1) Farthest point sampling: one workgroup per batch, dist[] lives in
//    LDS (256 KB of the 320 KB WGP LDS) across all 10 iterations.
// =====================================================================
__global__ void k_fps(const float* __restrict__ pc, float* __restrict__ centers) {
    extern __shared__ float sdist[];            // NPTS floats (256 KB)
    __shared__ float s_rv[32];
    __shared__ int   s_ri[32];
    __shared__ int   s_far;

    const int b   = blockIdx.x;
    const int tid = threadIdx.x;                // 0..1023
    const float* __restrict__ px = pc + (size_t)b * 3 * NPTS;
    const float* __restrict__ py = px + NPTS;
    const float* __restrict__ pz = px + 2 * NPTS;

    for (int n = tid; n < NPTS; n += 1024) sdist[n] = 1e10f;
    if (tid == 0) s_far = 0;
    __syncthreads();

    for (int it = 0; it < NPOINT; ++it) {
        const int far = s_far;
        const float cx = px[far], cy = py[far], cz = pz[far];
        if (tid == 0) {
            float* c = centers + (size_t)(b * NPOINT + it) * 3;
            c[0] = cx; c[1] = cy; c[2] = cz;
        }
        float bv = -1.0f; int bi = 0;
        for (int n = tid; n < NPTS; n += 1024) {
            // stream ahead (lowers to global_prefetch_b8)
            __builtin_prefetch(px + n + 8192, 0, 1);
            __builtin_prefetch(py + n + 8192, 0, 1);
            __builtin_prefetch(pz + n + 8192, 0, 1);
            const float dx = px[n] - cx, dy = py[n] - cy, dz = pz[n] - cz;
            const float d  = fminf(sdist[n], dx * dx + dy * dy + dz * dz);
            sdist[n] = d;
            if (d > bv) { bv = d; bi = n; }     // strict '>' keeps lowest index
        }
        // wave32 argmax reduction (tie -> lower index), then cross-wave via LDS
        for (int off = 16; off > 0; off >>= 1) {
            const float ov = __shfl_down(bv, off, 32);
            const int   oi = __shfl_down(bi, off, 32);
            if (ov > bv || (ov == bv && oi < bi)) { bv = ov; bi = oi; }
        }
        if ((tid & 31) == 0) { s_rv[tid >> 5] = bv; s_ri[tid >> 5] = bi; }
        __syncthreads();
        if (tid < 32) {
            bv = s_rv[tid]; bi = s_ri[tid];
            for (int off = 16; off > 0; off >>= 1) {
                const float ov = __shfl_down(bv, off, 32);
                const int   oi = __shfl_down(bi, off, 32);
                if (ov > bv || (ov == bv && oi < bi)) { bv = ov; bi = oi; }
            }
            if (tid == 0) s_far = bi;
        }
        __syncthreads();
    }
}

// =====================================================================
// 2) Ball query + group + center-subtract: one wave per (batch, center).
//    Index-ordered scan with wave32 ballot and early exit after 8 hits.
// =====================================================================
__global__ void k_ballq(const float* __restrict__ pc,
                        const float* __restrict__ centers,
                        float* __restrict__ grouped) {
    __shared__ int s_idx[8][NSAMPLE];
    const int wib  = threadIdx.x >> 5;
    const int lane = threadIdx.x & 31;
    const int w = blockIdx.x * 8 + wib;         // exactly BATCH*NPOINT waves
    const int b = w / NPOINT;
    const float* __restrict__ px = pc + (size_t)b * 3 * NPTS;
    const float* __restrict__ py = px + NPTS;
    const float* __restrict__ pz = px + 2 * NPTS;
    const float* c = centers + (size_t)w * 3;
    const float cx = c[0], cy = c[1], cz = c[2];

    int cnt = 0;
    for (int base = 0; base < NPTS; base += 32) {
        const int n = base + lane;
        const float dx = px[n] - cx, dy = py[n] - cy, dz = pz[n] - cz;
        const float d = dx * dx + dy * dy + dz * dz;
        unsigned mask = (unsigned)__ballot(d <= RAD2);   // wave32 -> low 32 bits
        while (mask && cnt < NSAMPLE) {
            const int bit = __builtin_ctz(mask);
            if (lane == 0) s_idx[wib][cnt] = base + bit;
            ++cnt;
            mask &= mask - 1;
        }
        if (cnt >= NSAMPLE) break;              // wave-uniform early exit
    }
    if (cnt == 0) { if (lane == 0) s_idx[wib][0] = 0; cnt = 1; }
    __syncthreads();
    if (lane < NSAMPLE) {
        const int gi = (lane < cnt) ? s_idx[wib][lane] : s_idx[wib][0];
        float* o = grouped + ((size_t)w * NSAMPLE + lane) * 3;
        o[0] = px[gi] - cx; o[1] = py[gi] - cy; o[2] = pz[gi] - cz;
    }
}

// =====================================================================
// 3) Tiny pointwise conv (oc,cin matmul) + channel sum/sumsq for BN.
//    layout flag: 1 -> input is grouped [b][s][n][c], 0 -> [b][c][n][s]
// =====================================================================
__global__ void k_pconv(const float* __restrict__ in, const float* __restrict__ w,
                        const float* __restrict__ bias, float* __restrict__ out,
                        float* __restrict__ sum, float* __restrict__ ssum,
                        int cin, int cout, int grouped_layout) {
    __shared__ float ls[16], lss[16];
    if (threadIdx.x < 16) { ls[threadIdx.x] = 0.f; lss[threadIdx.x] = 0.f; }
    __syncthreads();
    const int i = blockIdx.x * blockDim.x + threadIdx.x;
    const int total = BATCH * cout * NSAMPLE * NPOINT;
    if (i < total) {
        const int s = i % NPOINT;
        const int n = (i / NPOINT) % NSAMPLE;
        const int o = (i / (NPOINT * NSAMPLE)) % cout;
        const int b = i / (NPOINT * NSAMPLE * cout);
        float acc = bias[o];
        for (int c = 0; c < cin; ++c) {
            const float xv = grouped_layout
                ? in[((size_t)(b * NPOINT + s) * NSAMPLE + n) * 3 + c]
                : in[((size_t)(b * cin + c) * NSAMPLE + n) * NPOINT + s];
            acc += w[o * cin + c] * xv;
        }
        out[i] = acc;                            // layout [b][o][n][s] == i
        atomicAdd(&ls[o], acc);
        atomicAdd(&lss[o], acc * acc);
    }
    __syncthreads();
    if (threadIdx.x < (unsigned)cout) {
        atomicAdd(&sum[threadIdx.x],  ls[threadIdx.x]);
        atomicAdd(&ssum[threadIdx.x], lss[threadIdx.x]);
    }
}

// BN (population var) + ReLU, per-channel stats over B*8*10 = 10240 elems
__global__ void k_bnrelu(const float* __restrict__ t, const float* __restrict__ sum,
                         const float* __restrict__ ssum, const float* __restrict__ g,
                         const float* __restrict__ be, float* __restrict__ u, int C) {
    const int i = blockIdx.x * blockDim.x + threadIdx.x;
    if (i >= BATCH * C * NSAMPLE * NPOINT) return;
    const int o = (i / (NPOINT * NSAMPLE)) % C;
    const float inv = 1.0f / (BATCH * NSAMPLE * NPOINT);
    const float m  = sum[o] * inv;
    const float va = ssum[o] * inv - m * m;
    const float x  = (t[i] - m) * rsqrtf(va + EPSV) * g[o] + be[o];
    u[i] = fmaxf(x, 0.f);
}

// BN+ReLU on layer-3 output then max over the 8 samples -> feat[b][o*10+s]
__global__ void k_bn3max(const float* __restrict__ t3, const float* __restrict__ sum,
                         const float* __restrict__ ssum, const float* __restrict__ g,
                         const float* __restrict__ be, float* __restrict__ feat) {
    const int i = blockIdx.x * blockDim.x + threadIdx.x;
    if (i >= BATCH * 16 * NPOINT) return;
    const int s = i % NPOINT;
    const int o = (i / NPOINT) % 16;
    const int b = i / (NPOINT * 16);
    const float inv = 1.0f / (BATCH * NSAMPLE * NPOINT);
    const float m  = sum[o] * inv;
    const float va = ssum[o] * inv - m * m;
    const float sc = rsqrtf(va + EPSV) * g[o];
    const float bo = be[o];
    float best = 0.f;                            // ReLU output is >= 0
    for (int n = 0; n < NSAMPLE; ++n) {
        const float x = (t3[((size_t)(b * 16 + o) * NSAMPLE + n) * NPOINT + s] - m) * sc + bo;
        best = fmaxf(best, fmaxf(x, 0.f));
    }
    feat[b * 160 + o * NPOINT + s] = best;
}

// =====================================================================
// 4) fc1: h = feat(128x160) @ fc1_w^T(160x128) + b, via V_WMMA_F32_16X16X4_F32.
//    One 16x16 output tile per wave; BN-over-batch sums fused in epilogue.
// =====================================================================
__global__ void k_fc1_wmma(const float* __restrict__ feat, const float* __restrict__ w,
                           const float* __restrict__ bias, float* __restrict__ h,
                           float* __restrict__ fsum, float* __restrict__ fss) {
    const int wib  = threadIdx.x >> 5;
    const int lane = threadIdx.x & 31;
    const int tile = blockIdx.x * 8 + wib;       // 64 tiles (8x8)
    const int bt = tile >> 3, ft = tile & 7;
    const int half = lane >> 4, r = lane & 15;
    const int arow = bt * 16 + r;                // batch row
    const int col  = ft * 16 + r;                // output feature
    v8f acc = {};
    for (int k0 = 0; k0 < 160; k0 += 4) {
        v2f a, bb;
        // A 16x4 layout: v0 = K(0|2), v1 = K(1|3) for lane halves
        a[0]  = feat[arow * 160 + k0 + 2 * half + 0];
        a[1]  = feat[arow * 160 + k0 + 2 * half + 1];
        // B 4x16 layout: row K striped across lanes; B[k][n] = w[n][k]
        bb[0] = w[col * 160 + k0 + 2 * half + 0];
        bb[1] = w[col * 160 + k0 + 2 * half + 1];
        acc = __builtin_amdgcn_wmma_f32_16x16x4_f32(
                  false, a, false, bb, (short)0, acc, false, false);
    }
    const float bv = bias[col];
    float ps = 0.f, pss = 0.f;
#pragma unroll
    for (int j = 0; j < 8; ++j) {
        const int row = bt * 16 + j + 8 * half;  // f32 C/D VGPR layout
        const float v = acc[j] + bv;
        h[row * 128 + col] = v;
        ps += v; pss += v * v;
    }
    atomicAdd(&fsum[col], ps);
    atomicAdd(&fss[col], pss);
}

// =====================================================================
// 5) fc2: BN(batch)+ReLU applied on the fly to h, then
//    out = h'(128x128) @ fc2_w^T(128x25) + b, via WMMA (N padded to 32).
// =====================================================================
__global__ void k_fc2_wmma(const float* __restrict__ h, const float* __restrict__ fsum,
                           const float* __restrict__ fss, const float* __restrict__ bg,
                           const float* __restrict__ bb, const float* __restrict__ w2,
                           const float* __restrict__ b2, float* __restrict__ out) {
    const int wib  = threadIdx.x >> 5;           // 16 waves (block 512)
    const int lane = threadIdx.x & 31;
    const int bt = wib >> 1, nt = wib & 1;
    const int half = lane >> 4, r = lane & 15;
    const int arow = bt * 16 + r;
    const int ncol = nt * 16 + r;
    const bool nok = ncol < 25;
    v8f acc = {};
    for (int k0 = 0; k0 < 128; k0 += 4) {
        v2f a, bv;
#pragma unroll
        for (int j = 0; j < 2; ++j) {
            const int k = k0 + 2 * half + j;
            const float m  = fsum[k] * (1.0f / 128.0f);
            const float va = fss[k] * (1.0f / 128.0f) - m * m;
            const float u  = (h[arow * 128 + k] - m) * rsqrtf(va + EPSV) * bg[k] + bb[k];
            a[j]  = fmaxf(u, 0.f);
            bv[j] = nok ? w2[ncol * 128 + k] : 0.f;
        }
        acc = __builtin_amdgcn_wmma_f32_16x16x4_f32(
                  false, a, false, bv, (short)0, acc, false, false);
    }
    if (nok) {
        const float bias = b2[ncol];
#pragma unroll
        for (int j = 0; j < 8; ++j) {
            const int row = bt * 16 + j + 8 * half;
            out[row * 25 + ncol] = acc[j] + bias;
        }
    }
}

__global__ void k_zero(float* __restrict__ p, int n) {
    const int i = blockIdx.x * blockDim.x + threadIdx.x;
    if (i < n) p[i] = 0.f;
}

// =====================================================================
extern "C" void kernel_launch(void* const* d_in, const int* in_sizes, int n_in,
                              void* d_out, int out_size, void* d_ws, size_t ws_size,
                              hipStream_t stream) {
    const float* pc    = (const float*)d_in[0];
    const float* w1    = (const float*)d_in[1];
    const float* b1    = (const float*)d_in[2];
    const float* g1    = (const float*)d_in[3];
    const float* be1   = (const float*)d_in[4];
    const float* w2    = (const float*)d_in[5];
    const float* b2    = (const float*)d_in[6];
    const float* g2    = (const float*)d_in[7];
    const float* be2   = (const float*)d_in[8];
    const float* w3    = (const float*)d_in[9];
    const float* b3    = (const float*)d_in[10];
    const float* g3    = (const float*)d_in[11];
    const float* be3   = (const float*)d_in[12];
    const float* fc1_w = (const float*)d_in[13];
    const float* fc1_b = (const float*)d_in[14];
    const float* bn1_g = (const float*)d_in[15];
    const float* bn1_b = (const float*)d_in[16];
    const float* fc2_w = (const float*)d_in[17];
    const float* fc2_b = (const float*)d_in[18];
    float* out = (float*)d_out;
    float* ws  = (float*)d_ws;

    float* centers = ws + OFF_CENTERS;
    float* grouped = ws + OFF_GROUPED;
    float* t1 = ws + OFF_T1;  float* u1 = ws + OFF_U1;
    float* t2 = ws + OFF_T2;  float* u2 = ws + OFF_U2;
    float* t3 = ws + OFF_T3;
    float* feat = ws + OFF_FEAT;
    float* h    = ws + OFF_H;
    float* S    = ws + OFF_SUMS;
    float *s1 = S,      *ss1 = S + 8;
    float *s2 = S + 16, *ss2 = S + 24;
    float *s3 = S + 32, *ss3 = S + 48;
    float *fsum = S + 64, *fss = S + 192;

    // allow 256 KB dynamic LDS for the FPS kernel (host-side, capture-safe)
    hipFuncSetAttribute(reinterpret_cast<const void*>(k_fps),
                        hipFuncAttributeMaxDynamicSharedMemorySize,
                        NPTS * (int)sizeof(float));

    // ws is poisoned by the harness: zero the reduction state each call
    k_zero<<<2, 256, 0, stream>>>(S, SUMS_FLOATS);

    // heavy stages
    k_fps  <<<BATCH, 1024, NPTS * sizeof(float), stream>>>(pc, centers);
    k_ballq<<<(BATCH * NPOINT) / 8, 256, 0, stream>>>(pc, centers, grouped);

    // tiny point-MLP with batch-global BN
    k_pconv <<<(BATCH * 8  * 80 + 255) / 256, 256, 0, stream>>>(grouped, w1, b1, t1, s1, ss1, 3, 8, 1);
    k_bnrelu<<<(BATCH * 8  * 80 + 255) / 256, 256, 0, stream>>>(t1, s1, ss1, g1, be1, u1, 8);
    k_pconv <<<(BATCH * 8  * 80 + 255) / 256, 256, 0, stream>>>(u1, w2, b2, t2, s2, ss2, 8, 8, 0);
    k_bnrelu<<<(BATCH * 8  * 80 + 255) / 256, 256, 0, stream>>>(t2, s2, ss2, g2, be2, u2, 8);
    k_pconv <<<(BATCH * 16 * 80 + 255) / 256, 256, 0, stream>>>(u2, w3, b3, t3, s3, ss3, 8, 16, 0);
    k_bn3max<<<(BATCH * 16 * NPOINT + 255) / 256, 256, 0, stream>>>(t3, s3, ss3, g3, be3, feat);

    // WMMA fully-connected head
    k_fc1_wmma<<<8, 256, 0, stream>>>(feat, fc1_w, fc1_b, h, fsum, fss);
    k_fc2_wmma<<<1, 512, 0, stream>>>(h, fsum, fss, bn1_g, bn1_b, fc2_w, fc2_b, out);
}